// YoloAnchorFreeLoss_19559281066232
// MI455X (gfx1250) — compile-verified
//
#include <hip/hip_runtime.h>
#include <hip/hip_bf16.h>
#include <math.h>

typedef __attribute__((ext_vector_type(2))) float v2f;
typedef __attribute__((ext_vector_type(8))) float v8f;

// Full 32-lane (wave32) sum using one V_WMMA_F32_16X16X4_F32 with an all-ones
// B matrix. A-matrix 16x4 f32 layout (ISA 7.12.2): VGPR0 holds K=0 (lanes 0-15,
// M=lane) and K=2 (lanes 16-31, M=lane-16); VGPR1 (K=1/K=3) is zeroed.
// D[m][n] = sum_k A[m][k] (replicated over n), so summing the 8 C VGPRs per
// lane gives sum_{m=0..7} rowsum (low lanes) / sum_{m=8..15} (high lanes);
// xor-shuffle by 16 completes the 32-lane total in every lane.
__device__ __forceinline__ float wave_sum_wmma(float x) {
  v2f a; a.x = x;    a.y = 0.0f;
  v2f b; b.x = 1.0f; b.y = 1.0f;
  v8f c = {};
  c = __builtin_amdgcn_wmma_f32_16x16x4_f32(false, a, false, b, (short)0, c,
                                            false, false);
  float s = ((c[0] + c[1]) + (c[2] + c[3])) + ((c[4] + c[5]) + (c[6] + c[7]));
  return s + __shfl_xor(s, 16, 32);
}

__global__ void __launch_bounds__(256)
yolo_loss_stage1(const float* __restrict__ pred_dist,
                 const float* __restrict__ pred_bboxes,
                 const float* __restrict__ anchor_points,
                 const float* __restrict__ target_bboxes,
                 const float* __restrict__ target_scores,
                 float2* __restrict__ block_partials,
                 int N, int A) {
  const int i = blockIdx.x * blockDim.x + threadIdx.x;
  float li = 0.0f, ld = 0.0f;

  if (i < N) {
    const float eps = 1e-7f;
    // weight = sum over 80 class scores (exactly 0 for non-fg anchors)
    const float4* sp =
        reinterpret_cast<const float4*>(target_scores + (size_t)i * 80);
    float w = 0.0f;
#pragma unroll
    for (int k = 0; k < 20; ++k) {
      float4 v = sp[k];
      w += (v.x + v.y) + (v.z + v.w);
    }

    if (w != 0.0f) {
      const float4 pb = reinterpret_cast<const float4*>(pred_bboxes)[i];
      const float4 tb = reinterpret_cast<const float4*>(target_bboxes)[i];

      // ---- CIoU ----
      float w1 = pb.z - pb.x, h1 = pb.w - pb.y + eps;
      float w2 = tb.z - tb.x, h2 = tb.w - tb.y + eps;
      float iw = fmaxf(fminf(pb.z, tb.z) - fmaxf(pb.x, tb.x), 0.0f);
      float ih = fmaxf(fminf(pb.w, tb.w) - fmaxf(pb.y, tb.y), 0.0f);
      float inter = iw * ih;
      float uni = w1 * h1 + w2 * h2 - inter + eps;
      float iou = inter / uni;
      float cw = fmaxf(pb.z, tb.z) - fminf(pb.x, tb.x);
      float ch = fmaxf(pb.w, tb.w) - fminf(pb.y, tb.y);
      float c2 = cw * cw + ch * ch + eps;
      float dx = tb.x + tb.z - pb.x - pb.z;
      float dy = tb.y + tb.w - pb.y - pb.w;
      float rho2 = (dx * dx + dy * dy) * 0.25f;
      float dat = atanf(w2 / h2) - atanf(w1 / h1);
      float v = 0.40528473456935109f * dat * dat;  // 4/pi^2
      float alpha = v / (v - iou + (1.0f + eps));
      float ciou = iou - (rho2 / c2 + v * alpha);
      li = (1.0f - ciou) * w;

      // ---- DFL ----
      const int a = i % A;
      const float2 ap = reinterpret_cast<const float2*>(anchor_points)[a];
      float t4[4];
      t4[0] = ap.x - tb.x;
      t4[1] = ap.y - tb.y;
      t4[2] = tb.z - ap.x;
      t4[3] = tb.w - ap.y;

      float dfl = 0.0f;
#pragma unroll
      for (int s = 0; s < 4; ++s) {
        float tv = fminf(fmaxf(t4[s], 0.0f), 14.99f);
        int tl = (int)tv;  // tv >= 0, floor == trunc; tl in [0,14]
        int tr = tl + 1;   // tr in [1,15]
        float wr = tv - (float)tl;  // 1 - wl
        float wl = 1.0f - wr;

        const float4* lp = reinterpret_cast<const float4*>(
            pred_dist + (size_t)i * 64 + s * 16);
        float q[16];
#pragma unroll
        for (int k = 0; k < 4; ++k) {
          float4 qq = lp[k];
          q[4 * k + 0] = qq.x;
          q[4 * k + 1] = qq.y;
          q[4 * k + 2] = qq.z;
          q[4 * k + 3] = qq.w;
        }
        float m = q[0];
#pragma unroll
        for (int k = 1; k < 16; ++k) m = fmaxf(m, q[k]);
        float sum = 0.0f, xl = 0.0f, xr = 0.0f;
#pragma unroll
        for (int k = 0; k < 16; ++k) {
          sum += expf(q[k] - m);
          if (k == tl) xl = q[k];
          if (k == tr) xr = q[k];
        }
        float lse = m + logf(sum);
        dfl += (lse - xl) * wl + (lse - xr) * wr;
      }
      ld = dfl * 0.25f * w;
    }
  }

  // Block reduction: one WMMA per wave per quantity (EXEC is all-ones here),
  // then 8 wave totals through LDS.
  float wli = wave_sum_wmma(li);
  float wld = wave_sum_wmma(ld);

  __shared__ float s_li[8];
  __shared__ float s_ld[8];
  const int lane = threadIdx.x & 31;
  const int wid = threadIdx.x >> 5;
  if (lane == 0) {
    s_li[wid] = wli;
    s_ld[wid] = wld;
  }
  __syncthreads();
  if (threadIdx.x == 0) {
    float pa = 0.0f, pb2 = 0.0f;
    const int nw = blockDim.x >> 5;
    for (int j = 0; j < nw; ++j) {
      pa += s_li[j];
      pb2 += s_ld[j];
    }
    block_partials[blockIdx.x] = make_float2(pa, pb2);
  }
}

__global__ void __launch_bounds__(256)
yolo_loss_stage2(const float2* __restrict__ partials, int nblocks,
                 const float* __restrict__ tss_ptr, float* __restrict__ out) {
  float li = 0.0f, ld = 0.0f;
  for (int j = threadIdx.x; j < nblocks; j += blockDim.x) {
    float2 p = partials[j];
    li += p.x;
    ld += p.y;
  }
  float wli = wave_sum_wmma(li);
  float wld = wave_sum_wmma(ld);

  __shared__ float s_li[8];
  __shared__ float s_ld[8];
  const int lane = threadIdx.x & 31;
  const int wid = threadIdx.x >> 5;
  if (lane == 0) {
    s_li[wid] = wli;
    s_ld[wid] = wld;
  }
  __syncthreads();
  if (threadIdx.x == 0) {
    float a = 0.0f, b = 0.0f;
    const int nw = blockDim.x >> 5;
    for (int j = 0; j < nw; ++j) {
      a += s_li[j];
      b += s_ld[j];
    }
    const float tss = tss_ptr[0];  // already max(sum, 1) per reference setup
    out[0] = a / tss;
    out[1] = b / tss;
  }
}

extern "C" void kernel_launch(void* const* d_in, const int* in_sizes, int n_in,
                              void* d_out, int out_size, void* d_ws,
                              size_t ws_size, hipStream_t stream) {
  const float* pred_dist     = (const float*)d_in[0];
  const float* pred_bboxes   = (const float*)d_in[1];
  const float* anchor_points = (const float*)d_in[2];
  const float* target_bboxes = (const float*)d_in[3];
  const float* target_scores = (const float*)d_in[4];
  const float* tss           = (const float*)d_in[5];
  // d_in[6] (fg_mask) intentionally unused: target_scores already embeds it
  // exactly (zero rows where fg==0), so weight==0 reproduces the masking.

  const int N = in_sizes[0] / 64;  // B*A (pred_dist is [B,A,4*16])
  const int A = in_sizes[2] / 2;   // anchor_points is [A,2]
  (void)A;

  const int threads = 256;
  const int blocks = (N + threads - 1) / threads;
  float2* partials = (float2*)d_ws;  // blocks * 8 bytes (~8.4 KB)

  yolo_loss_stage1<<<blocks, threads, 0, stream>>>(
      pred_dist, pred_bboxes, anchor_points, target_bboxes, target_scores,
      partials, N, A);
  yolo_loss_stage2<<<1, 256, 0, stream>>>(partials, blocks, tss,
                                          (float*)d_out);
}